// Ts_cov_48902497632337
// MI455X (gfx1250) — compile-verified
//
#include <hip/hip_runtime.h>
#include <hip/hip_bf16.h>

// out[b, p(i,j), t] = (1/5) * sum_{k=t..t+4} xn[b,i,k]*xn[b,j,k]
// where xn is per-row z-normalized input. Each 16x16 tile of the 128x128
// Gram matrix (upper-triangular tiles only) is computed with
// V_WMMA_F32_16X16X4_F32, two per window (K=4 main + K-slot-0-only tail).

#define BATCH 1024
#define RROWS 128
#define TT    30
#define TO    26          // TT - 5 + 1
#define NPAIR 8256        // 128*129/2
#define XSTR  33          // padded row stride in LDS (30 data + 3 zeros)
#define PSTR  28          // staging row stride (26 data + pad, 8B-aligned)

typedef __attribute__((ext_vector_type(2))) float v2f;
typedef __attribute__((ext_vector_type(8))) float v8f;

__global__ __launch_bounds__(32)
void tscov_gram_kernel(const float* __restrict__ in, float* __restrict__ out) {
    const int blkId = blockIdx.x;   // 0..35 : upper-tri 16x16 tile index
    const int b     = blockIdx.y;   // 0..1023

    // tile index -> (mi, ni), mi <= ni, in 0..7
    int mi = 0, rem = blkId;
    while (rem >= 8 - mi) { rem -= 8 - mi; ++mi; }
    const int ni = mi + rem;

    __shared__ float xsA[16 * XSTR];
    __shared__ float xsB[16 * XSTR];
    __shared__ float stage[256 * PSTR];

    const int lane = threadIdx.x;   // 0..31
    const int half = lane >> 4;     // 0 -> A row group, 1 -> B row group
    const int l16  = lane & 15;

    // ---- Phase 1: normalize 32 rows into LDS, scaled by 1/(std*sqrt(5)) ----
    {
        const int grp = half ? ni : mi;
        const int row = grp * 16 + l16;
        // rows are 120 B apart -> 8B aligned: load as 15 x b64
        const float2* src2 = (const float2*)(in + ((size_t)b * RROWS + row) * TT);
        float vals[TT];
        float s = 0.f;
        #pragma unroll
        for (int q = 0; q < TT / 2; ++q) {
            float2 v = src2[q];
            vals[2 * q]     = v.x;
            vals[2 * q + 1] = v.y;
            s += v.x + v.y;
        }
        const float mean = s * (1.0f / TT);
        float s2 = 0.f;
        #pragma unroll
        for (int t = 0; t < TT; ++t) { float c = vals[t] - mean; vals[t] = c; s2 += c * c; }
        // 1/std (population) * 1/sqrt(5): folds the window /5 into the data.
        const float scale = rsqrtf(s2 * (1.0f / TT)) * 0.44721359549995793f;
        float* dst = (half ? xsB : xsA) + l16 * XSTR;
        #pragma unroll
        for (int t = 0; t < TT; ++t) dst[t] = vals[t] * scale;
        dst[30] = 0.f; dst[31] = 0.f; dst[32] = 0.f;   // K-padding slots
    }
    __syncthreads();

    // ---- Phase 2: 26 windows, 2 WMMAs each; stage D tiles as [pair][t] ----
    // fp32 A layout (ISA 7.12.2): lanes 0-15 hold K{0,1} in v{0,1},
    // lanes 16-31 hold K{2,3}. B assumed mirror of A (N across lanes);
    // Gram symmetry means A and B operands are built identically.
    const int   kOfs  = 2 * half;          // per-lane K base within operand
    const float maskA = half ? 0.f : 1.f;  // zeroes A's K2/K3 slots (no branch)
    const float* arow = xsA + l16 * XSTR;
    const float* brow = xsB + l16 * XSTR;
    float* srow = stage + (half * 128 + l16) * PSTR;   // + v*16*PSTR per D-VGPR

    // Modest unroll: overlap two windows without blowing past 256 VGPRs
    // (full unroll forced s_set_vgpr_msb + heavy v_mov traffic).
    #pragma clang loop unroll_count(2)
    for (int t = 0; t < TO; ++t) {
        v2f a, bb, a2, b2;
        a.x  = arow[t + kOfs];     a.y  = arow[t + kOfs + 1];
        bb.x = brow[t + kOfs];     bb.y = brow[t + kOfs + 1];
        // Tail WMMA: only K-slot 0 (time t+4) is live. A is zeroed in all
        // other slots (mask multiply), so B's K2/K3 content is irrelevant
        // and B needs NO masking -> fully uniform loads, no exec juggling.
        a2.x = arow[t + 4] * maskA;  a2.y = 0.f;
        b2.x = brow[t + 4];          b2.y = 0.f;

        v8f c = {};
        c = __builtin_amdgcn_wmma_f32_16x16x4_f32(false, a,  false, bb,
                                                  (short)0, c, false, false);
        c = __builtin_amdgcn_wmma_f32_16x16x4_f32(false, a2, false, b2,
                                                  (short)0, c, false, false);

        // D layout: VGPR v -> M = half*8 + v, N = l16.
        #pragma unroll
        for (int v = 0; v < 8; ++v)
            srow[v * 16 * PSTR + t] = c[v];
    }
    __syncthreads();

    // ---- Phase 3: coalesced writeback, 26 contiguous floats per pair ----
    const size_t outBase = (size_t)b * NPAIR;
    #pragma unroll
    for (int k = 0; k < 8; ++k) {
        const int pl = lane + 32 * k;          // local pair 0..255
        const int m  = pl >> 4, n = pl & 15;
        const int i  = mi * 16 + m;
        const int j  = ni * 16 + n;
        if (i <= j) {
            const int p = i * RROWS - (i * (i - 1)) / 2 + (j - i);
            float*       dst = out + (outBase + (size_t)p) * TO;
            const float* srp = stage + pl * PSTR;
            #pragma unroll
            for (int q = 0; q < 13; ++q)       // 13 x b64 = 26 floats
                ((float2*)dst)[q] = ((const float2*)srp)[q];
        }
    }
}

extern "C" void kernel_launch(void* const* d_in, const int* in_sizes, int n_in,
                              void* d_out, int out_size, void* d_ws, size_t ws_size,
                              hipStream_t stream) {
    const float* in = (const float*)d_in[0];
    float* out = (float*)d_out;
    dim3 grid(36, BATCH);
    tscov_gram_kernel<<<grid, 32, 0, stream>>>(in, out);
}